// KNN_WeightNet_25220047962568
// MI455X (gfx1250) — compile-verified
//
#include <hip/hip_runtime.h>
#include <hip/hip_bf16.h>

typedef float v2f __attribute__((ext_vector_type(2)));
typedef float v8f __attribute__((ext_vector_type(8)));

#define Bb 8
#define Nn 4096
#define Kk 32
#define Cc 64
#define NBN (Bb * Nn)          // 32768 (b,n) pairs
#define WAVES 8                 // waves per workgroup

static __device__ __forceinline__ v8f wmma4(v2f a, v2f b, v8f c) {
  // D = A(16x4 f32) * B(4x16 f32) + C(16x16 f32), full fp32 precision
  return __builtin_amdgcn_wmma_f32_16x16x4_f32(
      /*neg_a=*/false, a, /*neg_b=*/false, b,
      /*c_mod=*/(short)0, c, /*reuse_a=*/false, /*reuse_b=*/false);
}

// -------- Kernel 1: MLP -> logits[b][n][k] ---------------------------------
// One wave per (b,n): 32 rows (k) x 64 features. Layers 1,2 on WMMA f32,
// layers 3,4 per-lane scalar after an LDS re-layout.
__global__ __launch_bounds__(256) void k_logits(
    const float* __restrict__ F,   // [B,N,K,C]
    const float* __restrict__ W1, const float* __restrict__ b1,   // [32,64],[32]
    const float* __restrict__ W2, const float* __restrict__ b2,   // [16,32],[16]
    const float* __restrict__ W3, const float* __restrict__ b3,   // [8,16],[8]
    const float* __restrict__ W4, const float* __restrict__ b4,   // [1,8],[1]
    float* __restrict__ logits)    // [B*N*K]
{
  __shared__ float sX2[WAVES][32 * 32];  // layer1 output (+b1), row=k, col=m1
  __shared__ float sL2[WAVES][32 * 16];  // relu(layer2 output), row=k, col=m2

  const int w    = threadIdx.x >> 5;
  const int lane = threadIdx.x & 31;
  const int wid  = blockIdx.x * WAVES + w;     // (b,n) index, exact grid
  const int h    = lane >> 4;                  // lane half (selects K-pair)
  const int lr   = lane & 15;                  // row/col within 16

  const float* Fw = F + (size_t)wid * (Kk * Cc);

  // ---- Layer 1: Out[32x32] = X[32x64] * W1^T, two row-tiles x two col-tiles
  v8f a00 = {}, a01 = {}, a10 = {}, a11 = {};
#pragma unroll
  for (int s = 0; s < 16; ++s) {
    const int kf = 4 * s + 2 * h;  // this lane-half's K pair
    v2f aT0 = *(const v2f*)(Fw + (size_t)lr * Cc + kf);          // rows k=0..15
    v2f aT1 = *(const v2f*)(Fw + (size_t)(lr + 16) * Cc + kf);   // rows k=16..31
    v2f bC0 = *(const v2f*)(W1 + (size_t)lr * Cc + kf);          // out cols 0..15
    v2f bC1 = *(const v2f*)(W1 + (size_t)(lr + 16) * Cc + kf);   // out cols 16..31
    a00 = wmma4(aT0, bC0, a00);
    a01 = wmma4(aT0, bC1, a01);
    a10 = wmma4(aT1, bC0, a10);
    a11 = wmma4(aT1, bC1, a11);
  }

  // C-layout -> LDS [k][m1] with bias (no activation after layer 1)
  {
    const float bi0 = b1[lr];
    const float bi1 = b1[lr + 16];
#pragma unroll
    for (int r = 0; r < 8; ++r) {
      const int k0 = r + 8 * h;  // row within tile
      sX2[w][k0 * 32 + lr]              = a00[r] + bi0;
      sX2[w][k0 * 32 + lr + 16]         = a01[r] + bi1;
      sX2[w][(k0 + 16) * 32 + lr]       = a10[r] + bi0;
      sX2[w][(k0 + 16) * 32 + lr + 16]  = a11[r] + bi1;
    }
  }
  __builtin_amdgcn_wave_barrier();  // same-wave LDS: DS ops are in-order

  // ---- Layer 2: Out[32x16] = X2[32x32] * W2^T
  v8f c20 = {}, c21 = {};
#pragma unroll
  for (int s = 0; s < 8; ++s) {
    const int kf = 4 * s + 2 * h;
    v2f a0 = *(const v2f*)(&sX2[w][lr * 32 + kf]);
    v2f a1 = *(const v2f*)(&sX2[w][(lr + 16) * 32 + kf]);
    v2f bb = *(const v2f*)(W2 + (size_t)lr * 32 + kf);
    c20 = wmma4(a0, bb, c20);
    c21 = wmma4(a1, bb, c21);
  }

  // ReLU + bias, C-layout -> LDS [k][m2]
  {
    const float bi2 = b2[lr];
#pragma unroll
    for (int r = 0; r < 8; ++r) {
      const int k0 = r + 8 * h;
      sL2[w][k0 * 16 + lr]        = fmaxf(c20[r] + bi2, 0.0f);
      sL2[w][(k0 + 16) * 16 + lr] = fmaxf(c21[r] + bi2, 0.0f);
    }
  }
  __builtin_amdgcn_wave_barrier();

  // ---- Layers 3 & 4: one row (k = lane) per lane, scalar
  float x[16];
#pragma unroll
  for (int i = 0; i < 16; i += 4) {
    float4 v = *(const float4*)(&sL2[w][lane * 16 + i]);
    x[i + 0] = v.x; x[i + 1] = v.y; x[i + 2] = v.z; x[i + 3] = v.w;
  }
  float acc4 = b4[0];
#pragma unroll
  for (int j = 0; j < 8; ++j) {
    float aj = b3[j];
#pragma unroll
    for (int i = 0; i < 16; ++i) aj = fmaf(x[i], W3[j * 16 + i], aj);
    acc4 = fmaf(fmaxf(aj, 0.0f), W4[j], acc4);
  }
  const float logit = fmaxf(acc4, 0.0f);  // final ReLU (affects ordering!)
  logits[(size_t)wid * Kk + lane] = logit;
}

// -------- Kernel 2: logD[b][k] = log(sum_n exp(logit)) — deterministic -----
__global__ __launch_bounds__(256) void k_denom(const float* __restrict__ logits,
                                               float* __restrict__ logD) {
  __shared__ float red[256];
  const int bk = blockIdx.x;           // 0..255 = b*32+k
  const int b  = bk >> 5;
  const int k  = bk & 31;
  const int t  = threadIdx.x;
  const float* base = logits + ((size_t)b * Nn) * Kk + k;
  float s = 0.0f;
#pragma unroll
  for (int i = 0; i < Nn / 256; ++i)   // fixed assignment & order per thread
    s += expf(base[(size_t)(t + 256 * i) * Kk]);
  red[t] = s;
  __syncthreads();
#pragma unroll
  for (int off = 128; off > 0; off >>= 1) {   // fixed tree order
    if (t < off) red[t] += red[t + off];
    __syncthreads();
  }
  if (t == 0) logD[bk] = logf(red[0]);
}

// -------- Kernel 3: stable top-16 over K=32 per (b,n) ----------------------
__global__ __launch_bounds__(256) void k_topk(const float* __restrict__ logits,
                                              const float* __restrict__ logD,
                                              int* __restrict__ out) {
  const int w    = threadIdx.x >> 5;
  const int lane = threadIdx.x & 31;       // k
  const int wid  = blockIdx.x * WAVES + w; // (b,n)
  const int b    = wid >> 12;              // / 4096
  // probs ordering over k == ordering of logit - logsumexp_n
  const float sc = logits[(size_t)wid * Kk + lane] - logD[b * Kk + lane];
  int rank = 0;
#pragma unroll
  for (int j = 0; j < 32; ++j) {
    const float o = __shfl(sc, j, 32);
    rank += (o > sc) || (o == sc && j < lane);  // JAX top_k stable tie-break
  }
  if (rank < 16) out[(size_t)wid * 16 + rank] = lane;
}

extern "C" void kernel_launch(void* const* d_in, const int* in_sizes, int n_in,
                              void* d_out, int out_size, void* d_ws, size_t ws_size,
                              hipStream_t stream) {
  const float* F  = (const float*)d_in[0];
  const float* W1 = (const float*)d_in[1];
  const float* b1 = (const float*)d_in[2];
  const float* W2 = (const float*)d_in[3];
  const float* b2 = (const float*)d_in[4];
  const float* W3 = (const float*)d_in[5];
  const float* b3 = (const float*)d_in[6];
  const float* W4 = (const float*)d_in[7];
  const float* b4 = (const float*)d_in[8];
  // d_in[9] = topk (16), hardcoded

  float* ws_logits = (float*)d_ws;                       // B*N*K floats = 4 MB
  float* ws_logD   = ws_logits + (size_t)Bb * Nn * Kk;   // 256 floats

  k_logits<<<NBN / WAVES, 32 * WAVES, 0, stream>>>(F, W1, b1, W2, b2, W3, b3,
                                                   W4, b4, ws_logits);
  k_denom<<<Bb * Kk, 256, 0, stream>>>(ws_logits, ws_logD);
  k_topk<<<NBN / WAVES, 32 * WAVES, 0, stream>>>(ws_logits, ws_logD,
                                                 (int*)d_out);
}